// RefineCostVolume_21457656611413
// MI455X (gfx1250) — compile-verified
//
#include <hip/hip_runtime.h>
#include <math.h>

#define B_   2
#define C_   32
#define H_   80
#define W_   160
#define L_   (H_ * W_)      // 12800
#define NH_  4
#define HD_  8
#define NW_  121            // 11x11 window

typedef __attribute__((ext_vector_type(2))) float v2f;
typedef __attribute__((ext_vector_type(8))) float v8f;

// ---------------------------------------------------------------------------
// 3x3 conv (Cin=Cout=32, SAME zero pad) as implicit GEMM on the fp32 WMMA path.
//   M=32 (co), K=288 ordered k=j*32+ci (tap-major), N=pixels.
// One wave -> one 32x16 output tile (two v8f accumulators), 144 WMMA per wave.
// Boundary handling: unconditional load from border-clamped address, value
// multiplied by a 0/1 lane mask -> straight-line loads the scheduler can
// clause & pipeline (no divergent saveexec around each load).
// Weights staged in LDS as [j][ci/2][co][2]: one aligned ds_load_b64 per
// A-fragment per lane, bank-conflict-free, no repacking movs.
// mode 0: BN+ReLU epilogue; mode 1: ReLU.
// ---------------------------------------------------------------------------
__global__ void conv3x3_wmma_kernel(const float* __restrict__ in,   // [B,32,H,W]
                                    const float* __restrict__ wgt,  // [32,32,3,3]
                                    const float* __restrict__ gamma,
                                    const float* __restrict__ beta,
                                    const float* __restrict__ mean,
                                    const float* __restrict__ var,
                                    float* __restrict__ out,        // [B,32,H,W]
                                    int mode) {
  __shared__ float wsm[9 * 32 * 32];   // [j][ci/2][co][ci&1], 36 KB
  // wgt[((co*32)+ci)*9 + j] -> wsm[((j*16 + ci/2)*32 + co)*2 + (ci&1)]
  for (int i = threadIdx.x; i < 9 * 32 * 32; i += blockDim.x) {
    int j   = i % 9;
    int rem = i / 9;
    int ci  = rem % 32;
    int co  = rem / 32;
    wsm[((j * 16 + (ci >> 1)) * 32 + co) * 2 + (ci & 1)] = wgt[i];
  }
  __syncthreads();

  int gwave = (blockIdx.x * blockDim.x + threadIdx.x) >> 5;
  int lane  = threadIdx.x & 31;
  int tilesN = L_ / 16;                 // 800
  if (gwave >= B_ * tilesN) return;     // grid is exact; wave-uniform anyway
  int b  = gwave / tilesN;
  int tn = gwave % tilesN;
  int n0 = tn * 16;
  int y  = n0 / W_;                     // W_ % 16 == 0: tile lies in one image row
  int x0 = n0 % W_;
  int lh = lane & 15;
  int hi = lane >> 4;
  const float* inb = in + (size_t)b * C_ * L_;

  v8f acc0 = {0.f, 0.f, 0.f, 0.f, 0.f, 0.f, 0.f, 0.f};  // co 0..15
  v8f acc1 = {0.f, 0.f, 0.f, 0.f, 0.f, 0.f, 0.f, 0.f};  // co 16..31

  #pragma unroll
  for (int j = 0; j < 9; ++j) {
    int ky = j / 3, kx = j % 3;
    int yy = y + ky - 1;
    int xx = x0 + lh + kx - 1;                       // lane-dependent
    bool ok = (yy >= 0) & (yy < H_) & (xx >= 0) & (xx < W_);
    float okf = ok ? 1.f : 0.f;
    int yyc = yy < 0 ? 0 : (yy >= H_ ? H_ - 1 : yy); // border-clamped (safe) addr
    int xxc = xx < 0 ? 0 : (xx >= W_ ? W_ - 1 : xx);
    size_t boff = (size_t)yyc * W_ + xxc;
    const float* wj = wsm + j * 1024;
    #pragma unroll
    for (int kk = 0; kk < 32; kk += 4) {
      int ci     = kk + hi * 2;
      int cihalf = (kk >> 1) + hi;                   // ci/2 (kk even)
      v2f bv;
      bv.x = inb[(size_t)ci * L_ + boff] * okf;
      bv.y = inb[(size_t)(ci + 1) * L_ + boff] * okf;
      v2f a0 = *(const v2f*)(wj + (cihalf * 32 + lh) * 2);
      v2f a1 = *(const v2f*)(wj + (cihalf * 32 + 16 + lh) * 2);
      acc0 = __builtin_amdgcn_wmma_f32_16x16x4_f32(
          false, a0, false, bv, (short)0, acc0, false, false);
      acc1 = __builtin_amdgcn_wmma_f32_16x16x4_f32(
          false, a1, false, bv, (short)0, acc1, false, false);
    }
  }

  int col = n0 + lh;
  float* ob = out + (size_t)b * C_ * L_;
  #pragma unroll
  for (int r = 0; r < 8; ++r) {
    int co0 = r + hi * 8;
    int co1 = co0 + 16;
    float s0 = acc0[r];
    float s1 = acc1[r];
    if (mode == 0) {
      float sc0 = rsqrtf(var[co0] + 1e-5f) * gamma[co0];
      float sc1 = rsqrtf(var[co1] + 1e-5f) * gamma[co1];
      s0 = (s0 - mean[co0]) * sc0 + beta[co0];
      s1 = (s1 - mean[co1]) * sc1 + beta[co1];
    }
    ob[(size_t)co0 * L_ + col] = fmaxf(s0, 0.f);
    ob[(size_t)co1 * L_ + col] = fmaxf(s1, 0.f);
  }
}

// 3x3 conv, Cin=32, Cout=1, sigmoid epilogue (mask prediction). Tiny: VALU.
__global__ void conv3x3_sig_kernel(const float* __restrict__ in,
                                   const float* __restrict__ wgt,
                                   float* __restrict__ out) {
  int idx = blockIdx.x * blockDim.x + threadIdx.x;
  if (idx >= B_ * L_) return;
  int x = idx % W_;
  int y = (idx / W_) % H_;
  int b = idx / L_;
  const float* inb = in + (size_t)b * C_ * L_;
  float s = 0.f;
  #pragma unroll 4
  for (int ci = 0; ci < C_; ++ci) {
    const float* ip = inb + (size_t)ci * L_;
    const float* wp = wgt + ci * 9;
    #pragma unroll
    for (int ky = 0; ky < 3; ++ky) {
      int yy = y + ky - 1;
      if (yy < 0 || yy >= H_) continue;
      #pragma unroll
      for (int kx = 0; kx < 3; ++kx) {
        int xx = x + kx - 1;
        if (xx < 0 || xx >= W_) continue;
        s = fmaf(ip[yy * W_ + xx], wp[ky * 3 + kx], s);
      }
    }
  }
  out[idx] = 1.f / (1.f + __expf(-s));
}

// ---------------------------------------------------------------------------
// fp32 WMMA GEMM: O[b,M,L] = A[M,32] @ X[b,32,L]   (1x1 qkv conv, M=96)
// ---------------------------------------------------------------------------
__global__ void gemm32_wmma_kernel(const float* __restrict__ A,
                                   const float* __restrict__ X,
                                   float* __restrict__ O,
                                   int M) {
  int gwave = (blockIdx.x * blockDim.x + threadIdx.x) >> 5;
  int lane  = threadIdx.x & 31;
  int tilesN = L_ / 16;            // 800
  int tilesM = M / 16;
  int tilesPerB = tilesM * tilesN;
  if (gwave >= B_ * tilesPerB) return;   // wave-uniform: EXEC all-ones for WMMA
  int b  = gwave / tilesPerB;
  int t  = gwave % tilesPerB;
  int tm = t / tilesN, tn = t % tilesN;
  int m0 = tm * 16, n0 = tn * 16;
  int lh = lane & 15;
  int hi = lane >> 4;
  const float* Xb = X + (size_t)b * C_ * L_;
  v8f acc = {0.f, 0.f, 0.f, 0.f, 0.f, 0.f, 0.f, 0.f};
  #pragma unroll
  for (int kk = 0; kk < 32; kk += 4) {
    int ka = kk + hi * 2;
    v2f av, bv;
    av.x = A[(m0 + lh) * 32 + ka];
    av.y = A[(m0 + lh) * 32 + ka + 1];
    bv.x = Xb[(size_t)ka * L_ + n0 + lh];
    bv.y = Xb[(size_t)(ka + 1) * L_ + n0 + lh];
    acc = __builtin_amdgcn_wmma_f32_16x16x4_f32(
        false, av, false, bv, (short)0, acc, false, false);
  }
  float* Ob = O + (size_t)b * M * L_;
  int col = n0 + lh;
  int rb  = m0 + hi * 8;
  #pragma unroll
  for (int r = 0; r < 8; ++r)
    Ob[(size_t)(rb + r) * L_ + col] = acc[r];
}

// ---------------------------------------------------------------------------
// 11x11 masked local-window attention, online softmax. One thread per
// (batch, head, pixel). Masked/OOB slots contribute exp(pos_bias) to the
// denominator but zero to the value aggregation (matches unfold semantics).
// ---------------------------------------------------------------------------
__global__ void attn_kernel(const float* __restrict__ qkv,      // [B,96,L]
                            const float* __restrict__ mpred,    // [B,L]
                            const float* __restrict__ pos_bias, // [NH,121]
                            float* __restrict__ outbuf) {       // [B,32,L]
  int idx = blockIdx.x * blockDim.x + threadIdx.x;
  if (idx >= B_ * NH_ * L_) return;
  int l  = idx % L_;
  int bh = idx / L_;
  int h  = bh % NH_;
  int b  = bh / NH_;
  int y = l / W_, x = l % W_;
  const float* qb = qkv + ((size_t)b * 96 + h * HD_) * L_;
  const float* kb = qkv + ((size_t)b * 96 + 32 + h * HD_) * L_;
  const float* vb = qkv + ((size_t)b * 96 + 64 + h * HD_) * L_;
  const float* mp = mpred + (size_t)b * L_;
  const float* pbh = pos_bias + h * NW_;

  float q[HD_];
  #pragma unroll
  for (int d = 0; d < HD_; ++d) q[d] = qb[(size_t)d * L_ + l];

  const float scale = 0.35355339059327373f;  // 8^-0.5
  float mmax = -1e30f, ssum = 0.f;
  float acc[HD_];
  #pragma unroll
  for (int d = 0; d < HD_; ++d) acc[d] = 0.f;

  int widx = 0;
  for (int di = -5; di <= 5; ++di) {
    int yy = y + di;
    bool yok = (yy >= 0) & (yy < H_);
    for (int dj = -5; dj <= 5; ++dj, ++widx) {
      int xx = x + dj;
      bool valid = yok & (xx >= 0) & (xx < W_);
      int ll = valid ? (yy * W_ + xx) : 0;
      float wm = valid ? ((mp[ll] > 0.5f) ? 1.f : 0.f) : 0.f;
      float logit = pbh[widx];
      if (wm != 0.f) {
        float dot = 0.f;
        #pragma unroll
        for (int d = 0; d < HD_; ++d)
          dot = fmaf(q[d], kb[(size_t)d * L_ + ll], dot);
        logit = fmaf(dot, scale, logit);
      }
      float nm = fmaxf(mmax, logit);
      float cs = __expf(mmax - nm);
      float p  = __expf(logit - nm);
      ssum = ssum * cs + p;
      if (wm != 0.f) {
        #pragma unroll
        for (int d = 0; d < HD_; ++d)
          acc[d] = fmaf(p, vb[(size_t)d * L_ + ll], acc[d] * cs);
      } else {
        #pragma unroll
        for (int d = 0; d < HD_; ++d) acc[d] *= cs;
      }
      mmax = nm;
    }
  }
  float mlow = (mp[l] > 0.5f) ? 0.f : 1.f;
  float inv  = mlow / ssum;
  float* ob = outbuf + ((size_t)b * C_ + h * HD_) * L_ + l;
  #pragma unroll
  for (int d = 0; d < HD_; ++d) ob[(size_t)d * L_] = acc[d] * inv;
}

// ---------------------------------------------------------------------------
// fp32 WMMA GEMM (M=32) + fused residual epilogue:
//   out[b,c,l] = feat[b,c,l] + (Wp @ Xattn + bias)[c,l] * mask_low[b,l]
// ---------------------------------------------------------------------------
__global__ void proj_wmma_kernel(const float* __restrict__ Wp,   // [32,32]
                                 const float* __restrict__ Xa,   // [B,32,L]
                                 const float* __restrict__ bias, // [32]
                                 const float* __restrict__ feat, // [B,32,L]
                                 const float* __restrict__ mpred,// [B,L]
                                 float* __restrict__ out) {      // [B,32,L]
  int gwave = (blockIdx.x * blockDim.x + threadIdx.x) >> 5;
  int lane  = threadIdx.x & 31;
  int tilesN = L_ / 16;                  // 800
  int tilesPerB = 2 * tilesN;            // M=32 -> 2 row tiles
  if (gwave >= B_ * tilesPerB) return;
  int b  = gwave / tilesPerB;
  int t  = gwave % tilesPerB;
  int tm = t / tilesN, tn = t % tilesN;
  int m0 = tm * 16, n0 = tn * 16;
  int lh = lane & 15;
  int hi = lane >> 4;
  const float* Xb = Xa + (size_t)b * C_ * L_;
  v8f acc = {0.f, 0.f, 0.f, 0.f, 0.f, 0.f, 0.f, 0.f};
  #pragma unroll
  for (int kk = 0; kk < 32; kk += 4) {
    int ka = kk + hi * 2;
    v2f av, bv;
    av.x = Wp[(m0 + lh) * 32 + ka];
    av.y = Wp[(m0 + lh) * 32 + ka + 1];
    bv.x = Xb[(size_t)ka * L_ + n0 + lh];
    bv.y = Xb[(size_t)(ka + 1) * L_ + n0 + lh];
    acc = __builtin_amdgcn_wmma_f32_16x16x4_f32(
        false, av, false, bv, (short)0, acc, false, false);
  }
  int col = n0 + lh;
  float ml = (mpred[(size_t)b * L_ + col] > 0.5f) ? 0.f : 1.f;
  int rb = m0 + hi * 8;
  const float* fb = feat + (size_t)b * C_ * L_;
  float* ob = out + (size_t)b * C_ * L_;
  #pragma unroll
  for (int r = 0; r < 8; ++r) {
    size_t o = (size_t)(rb + r) * L_ + col;
    ob[o] = fb[o] + (acc[r] + bias[rb + r]) * ml;
  }
}

// ---------------------------------------------------------------------------
extern "C" void kernel_launch(void* const* d_in, const int* in_sizes, int n_in,
                              void* d_out, int out_size, void* d_ws, size_t ws_size,
                              hipStream_t stream) {
  const float* featL = (const float*)d_in[0];
  const float* featR = (const float*)d_in[1];
  const float* w1    = (const float*)d_in[2];
  const float* gamma = (const float*)d_in[3];
  const float* beta  = (const float*)d_in[4];
  const float* mean  = (const float*)d_in[5];
  const float* var   = (const float*)d_in[6];
  const float* w2    = (const float*)d_in[7];
  const float* w3    = (const float*)d_in[8];
  const float* qkvw  = (const float*)d_in[9];
  const float* projw = (const float*)d_in[10];
  const float* projb = (const float*)d_in[11];
  const float* posb  = (const float*)d_in[12];
  float* out = (float*)d_out;
  float* ws  = (float*)d_ws;

  const int FEAT = B_ * C_ * L_;     // 819200
  const int MASK = B_ * L_;          // 25600
  float* h1   = ws;                  // [B,32,L]
  float* h2   = ws + FEAT;           // [B,32,L]
  float* qkv  = ws + 2 * FEAT;       // [B,96,L]
  float* aout = qkv + 3 * FEAT;      // [B,32,L]

  // conv (implicit GEMM): exactly B_*(L_/16)=1600 waves -> 200 blocks x 8 waves
  const int convBlocks = (B_ * (L_ / 16) * 32) / 256;   // 200

  for (int s = 0; s < 2; ++s) {
    const float* feat = s ? featR : featL;
    float* mpred = out + 2 * FEAT + s * MASK;  // mask_pred_{L,R}
    float* fref  = out + s * FEAT;             // feat{L,R}_ref

    conv3x3_wmma_kernel<<<convBlocks, 256, 0, stream>>>(
        feat, w1, gamma, beta, mean, var, h1, 0);
    conv3x3_wmma_kernel<<<convBlocks, 256, 0, stream>>>(
        h1, w2, nullptr, nullptr, nullptr, nullptr, h2, 1);
    conv3x3_sig_kernel<<<(MASK + 255) / 256, 256, 0, stream>>>(h2, w3, mpred);

    int qkvWaves = B_ * (96 / 16) * (L_ / 16);        // 9600 waves
    gemm32_wmma_kernel<<<(qkvWaves * 32) / 256, 256, 0, stream>>>(
        qkvw, feat, qkv, 96);

    int attnThreads = B_ * NH_ * L_;                  // 102400
    attn_kernel<<<(attnThreads + 255) / 256, 256, 0, stream>>>(
        qkv, mpred, posb, aout);

    int projWaves = B_ * 2 * (L_ / 16);               // 3200 waves
    proj_wmma_kernel<<<(projWaves * 32) / 256, 256, 0, stream>>>(
        projw, aout, projb, feat, mpred, fref);
  }
}